// SVM_42941083025647
// MI455X (gfx1250) — compile-verified
//
#include <hip/hip_runtime.h>

// Fused multiclass-SVM hinge gradient for MI455X (gfx1250, wave32).
//
// Roofline: X = 1M x 128 fp32 = 512 MB is the only large operand; at
// 23.3 TB/s HBM the floor is ~22us iff X is streamed exactly once.  The two
// GEMM pieces (scores = X.wT and grad += coefT.X) total ~5.1 GFLOP ->
// ~233 TFLOP/s fp32 needed to stay HBM-bound, which requires the matrix
// pipe: V_WMMA_F32_16X16X4_F32 (full fp32, matches reference numerics).
//
// Data movement (CDNA5 async path): each wave double-buffers 16x128 fp32
// tiles in LDS and stages them with GLOBAL_LOAD_ASYNC_TO_LDS_B128
// (ASYNCcnt; in-order completion), so the next tile's HBM fetch overlaps
// the current tile's 64 WMMAs.  s_wait_asynccnt 0x10 = "older tile done".
// w (B-layout) and the [16x128] fp32 gradient accumulators stay register-
// resident; reduction is ds_add_f32 (block) -> global_atomic_add_f32
// (device, raw sums) -> tiny epilogue applies 1/N and regularization.

#define SVM_REG   0.05f
#define SVM_D     128
#define SVM_C     10
#define TILE_M    16
#define TILE_F    (TILE_M * SVM_D)  // 2048 floats = 8 KB
#define WAVES_PB  4                 // waves per block (wave32)
#define MAIN_BLK  (WAVES_PB * 32)   // 128 threads
#define MAIN_GRID 512

typedef __attribute__((ext_vector_type(2))) float v2f;
typedef __attribute__((ext_vector_type(8))) float v8f;

static __device__ __forceinline__ v8f wmma_f32_4(v2f a, v2f b, v8f c) {
  // 8 args: (neg_a, A, neg_b, B, c_mod, C, reuse_a, reuse_b)
  return __builtin_amdgcn_wmma_f32_16x16x4_f32(false, a, false, b, (short)0, c,
                                               false, false);
}

// Issue one 16x128 fp32 tile (8 KB, contiguous in row-major X) as 16
// GLOBAL_LOAD_ASYNC_TO_LDS_B128 ops: each lane moves 16B per op.
static __device__ __forceinline__ void async_tile(const float* __restrict__ X,
                                                  size_t rowBase,
                                                  unsigned ldsOff, int lane) {
  unsigned long long g =
      (unsigned long long)(const void*)(X + rowBase * SVM_D) +
      (unsigned long long)(lane * 16);
  unsigned l = ldsOff + (unsigned)(lane * 16);
#pragma unroll
  for (int i = 0; i < 16; ++i) {
    unsigned li = l + (unsigned)(i * 512);
    unsigned long long gi = g + (unsigned long long)(i * 512);
    asm volatile("global_load_async_to_lds_b128 %0, %1, off"
                 :: "v"(li), "v"(gi)
                 : "memory");
  }
}

__global__ void __launch_bounds__(256) svm_zero(float* __restrict__ out, int n) {
  int i = blockIdx.x * blockDim.x + threadIdx.x;
  if (i < n) out[i] = 0.0f;
}

__global__ void __launch_bounds__(MAIN_BLK)
svm_main(const float* __restrict__ X, const int* __restrict__ y,
         const float* __restrict__ w, float* __restrict__ out, int numTiles) {
  // 4 waves x 2 tile buffers x 8 KB = 64 KB.  After the streaming loop the
  // same memory is re-used (barrier-protected) as the block-level gradient
  // accumulator (first 2048 floats) + loss cell (smem[2048]).
  __shared__ __align__(16) float smem[WAVES_PB * 2 * TILE_F];

  const int tid  = threadIdx.x;
  const int lane = tid & 31;
  const int wave = tid >> 5;
  const int l15  = lane & 15;   // class index (scores D) / column index
  const int hi   = lane >> 4;   // half-wave select
  float* tbuf0 = &smem[wave * (2 * TILE_F)];
  const unsigned ldsOff0 = (unsigned)(size_t)(void*)tbuf0;  // LDS byte offset

  const int waveId     = blockIdx.x * WAVES_PB + wave;
  const int waveStride = gridDim.x * WAVES_PB;

  // Kick off the first tile fetch ASAP so it overlaps w staging below.
  if (waveId < numTiles)
    async_tile(X, (size_t)waveId * TILE_M, ldsOff0, lane);

  // w in WMMA B-layout for the scores GEMM: per K-chunk k, lane holds
  // wT(K=4k+2*hi + {0,1}, N=l15) = w[l15][col], classes >=10 padded with 0.
  v2f wb[32];
#pragma unroll
  for (int k = 0; k < 32; ++k) {
    const int col = 4 * k + 2 * hi;
    float b0 = 0.0f, b1 = 0.0f;
    if (l15 < SVM_C) {
      b0 = w[l15 * SVM_D + col];
      b1 = w[l15 * SVM_D + col + 1];
    }
    wb[k].x = b0;
    wb[k].y = b1;
  }

  v8f gacc[8]; // grad accumulators: 8 D-slices of 16 (classes) x 16 (cols)
#pragma unroll
  for (int n = 0; n < 8; ++n)
    gacc[n] = v8f{0.f, 0.f, 0.f, 0.f, 0.f, 0.f, 0.f, 0.f};
  float lossAcc = 0.0f;

  int cur = 0;
  for (int t = waveId; t < numTiles; t += waveStride) {
    const size_t rowBase = (size_t)t * TILE_M;
    const int tn = t + waveStride;

    // Prefetch next tile into the other buffer, then wait for the current
    // one.  Async loads complete in order: <=16 outstanding => tile t done.
    if (tn < numTiles) {
      async_tile(X, (size_t)tn * TILE_M, ldsOff0 + (unsigned)((cur ^ 1) * (TILE_F * 4)), lane);
      const int t2 = tn + waveStride;   // warm L2 one tile further out
      if (t2 < numTiles)
        __builtin_prefetch(X + (size_t)t2 * TILE_M * SVM_D, 0, 1);
      asm volatile("s_wait_asynccnt 0x10" ::: "memory");
    } else {
      asm volatile("s_wait_asynccnt 0x0" ::: "memory");
    }
    const float* tile = tbuf0 + cur * TILE_F;

    // ---- scores = Xtile @ wT : 32x V_WMMA_F32_16X16X4_F32 over K=128 ----
    v8f sc = v8f{0.f, 0.f, 0.f, 0.f, 0.f, 0.f, 0.f, 0.f};
#pragma unroll
    for (int k = 0; k < 32; ++k) {
      // A-layout: lane<16 -> (M=l15, K=4k,4k+1); lane>=16 -> (M=l15, K=4k+2,4k+3)
      const v2f a =
          *reinterpret_cast<const v2f*>(&tile[l15 * SVM_D + 4 * k + 2 * hi]);
      sc = wmma_f32_4(a, wb[k], sc);
    }

    // ---- hinge: margins, mask, counts, coef (all cross-lane in D-layout) ----
    const int ylane = y[rowBase + l15]; // lane l holds y of tile-row l%16
    float maskv[8], coefv[8];
#pragma unroll
    for (int g = 0; g < 8; ++g) {
      const int ylow   = __shfl(ylane, g, 32);
      const int yhighv = __shfl(ylane, g + 8, 32);
      const int yrow   = hi ? yhighv : ylow; // correct class of this lane's row
      const float s    = sc[g];
      const float corr = __shfl(s, (hi << 4) | yrow, 32);
      const float margin = s - corr + 1.0f;
      const float m =
          (margin > 0.0f && l15 != yrow && l15 < SVM_C) ? 1.0f : 0.0f;
      maskv[g] = m;
      lossAcc += m * margin;
      // counts over the 16 classes: xor 1/2/4/8 stays within each half-wave
      float cnt = m;
      cnt += __shfl_xor(cnt, 1, 32);
      cnt += __shfl_xor(cnt, 2, 32);
      cnt += __shfl_xor(cnt, 4, 32);
      cnt += __shfl_xor(cnt, 8, 32);
      coefv[g] = (l15 == yrow) ? cnt : -m;
    }

    // ---- transpose coef (D-layout) into A-layout for grad GEMM ----
    // A chunk j needs rows K'={4j..4j+3}: vgpr0 = rows 4j (lo) / 4j+2 (hi),
    // vgpr1 = rows 4j+1 (lo) / 4j+3 (hi).
    v2f ga[4];
    {
      const float s2 = __shfl(coefv[2], l15, 32);
      const float s3 = __shfl(coefv[3], l15, 32);
      ga[0].x = hi ? s2 : coefv[0];
      ga[0].y = hi ? s3 : coefv[1];
      const float s6 = __shfl(coefv[6], l15, 32);
      const float s7 = __shfl(coefv[7], l15, 32);
      ga[1].x = hi ? s6 : coefv[4];
      ga[1].y = hi ? s7 : coefv[5];
      const float u0 = __shfl(coefv[0], l15 + 16, 32);
      const float u1 = __shfl(coefv[1], l15 + 16, 32);
      ga[2].x = hi ? coefv[2] : u0;
      ga[2].y = hi ? coefv[3] : u1;
      const float u4 = __shfl(coefv[4], l15 + 16, 32);
      const float u5 = __shfl(coefv[5], l15 + 16, 32);
      ga[3].x = hi ? coefv[6] : u4;
      ga[3].y = hi ? coefv[7] : u5;
    }

    // ---- grad += coefT @ Xtile : 8 D-slices x 4 K'-chunks = 32 WMMAs ----
#pragma unroll
    for (int n = 0; n < 8; ++n) {
#pragma unroll
      for (int j = 0; j < 4; ++j) {
        const int r = 4 * j + 2 * hi; // B-layout row for this half-wave
        v2f b;
        b.x = tile[r * SVM_D + 16 * n + l15];
        b.y = tile[(r + 1) * SVM_D + 16 * n + l15];
        gacc[n] = wmma_f32_4(ga[j], b, gacc[n]);
      }
    }
    cur ^= 1;
  }

  // ---- re-use tile LDS as the block gradient accumulator ----
  __syncthreads();                     // everyone done streaming
  for (int i = tid; i < 16 * SVM_D; i += MAIN_BLK) smem[i] = 0.0f;
  if (tid == 0) smem[16 * SVM_D] = 0.0f;   // loss cell
  __syncthreads();

  // wave -> block reduction (LDS fp32 atomics)
#pragma unroll
  for (int n = 0; n < 8; ++n) {
#pragma unroll
    for (int g = 0; g < 8; ++g) {
      const int cls = g + 8 * hi; // D-layout: vgpr g holds classes g / g+8
      atomicAdd(&smem[cls * SVM_D + 16 * n + l15], gacc[n][g]);
    }
  }
  float l = lossAcc;
  l += __shfl_xor(l, 16, 32);
  l += __shfl_xor(l, 8, 32);
  l += __shfl_xor(l, 4, 32);
  l += __shfl_xor(l, 2, 32);
  l += __shfl_xor(l, 1, 32);
  if (lane == 0) atomicAdd(&smem[16 * SVM_D], l);
  __syncthreads();

  // block -> device reduction (raw sums; scaled in epilogue)
  for (int i = tid; i < SVM_C * SVM_D; i += MAIN_BLK)
    atomicAdd(&out[i], smem[i]); // classes 0..9 are the first 1280 slots
  if (tid == 0) atomicAdd(&out[SVM_C * SVM_D], smem[16 * SVM_D]);
}

__global__ void __launch_bounds__(256)
svm_epilogue(const float* __restrict__ w, float* __restrict__ out, float invN) {
  __shared__ float red[256];
  const int tid = threadIdx.x;
  float s = 0.0f;
  for (int i = tid; i < SVM_C * SVM_D; i += 256) {
    const float v = w[i];
    s += v * v;
  }
  red[tid] = s;
  __syncthreads();
  for (int off = 128; off > 0; off >>= 1) {
    if (tid < off) red[tid] += red[tid + off];
    __syncthreads();
  }
  for (int i = tid; i < SVM_C * SVM_D; i += 256)
    out[i] = out[i] * invN + SVM_REG * w[i];
  if (tid == 0)
    out[SVM_C * SVM_D] = out[SVM_C * SVM_D] * invN + 0.5f * SVM_REG * red[0];
}

extern "C" void kernel_launch(void* const* d_in, const int* in_sizes, int n_in,
                              void* d_out, int out_size, void* d_ws,
                              size_t ws_size, hipStream_t stream) {
  const float* X = (const float*)d_in[0];
  const int* y   = (const int*)d_in[1];
  const float* w = (const float*)d_in[2];
  float* out     = (float*)d_out;

  const int N        = in_sizes[1];       // 1,000,000 (divisible by 16)
  const int numTiles = N / TILE_M;
  const float invN   = 1.0f / (float)N;
  const int zeroN    = SVM_C * SVM_D + 1; // 1280 grads + 1 loss

  svm_zero<<<(zeroN + 255) / 256, 256, 0, stream>>>(out, zeroN);
  svm_main<<<MAIN_GRID, MAIN_BLK, 0, stream>>>(X, y, w, out, numTiles);
  svm_epilogue<<<1, 256, 0, stream>>>(w, out, invN);
}